// Mamba_76330158784653
// MI455X (gfx1250) — compile-verified
//
#include <hip/hip_runtime.h>
#include <hip/hip_bf16.h>

// ---------------- problem constants ----------------
#define D_MODEL   1024
#define D_STATE   16
#define D_CONV    4
#define D_INNER   2048
#define DT_RANK   64
#define BATCH     4
#define SEQLEN    2048
#define BL        (BATCH * SEQLEN)        // 8192 rows
#define XDBL_N    (DT_RANK + 2 * D_STATE) // 96

typedef __attribute__((ext_vector_type(2))) float v2f;
typedef __attribute__((ext_vector_type(8))) float v8f;

// =====================================================================
// Weight transpose:  Bt[n*K + k] = B[k*N + n]   (one-time, L2-resident)
// =====================================================================
__global__ __launch_bounds__(256)
void transpose_kernel(const float* __restrict__ in, float* __restrict__ out,
                      int K, int N) {
    const int idx = blockIdx.x * blockDim.x + threadIdx.x;   // over K*N
    if (idx >= K * N) return;
    const int k = idx / N;
    const int n = idx - k * N;
    out[(size_t)n * K + k] = in[idx];
}

// =====================================================================
// fp32 WMMA GEMM with transposed B:  C[M x N] = A[M x K(lda)] * Bt^T
//   Bt is N x K row-major, so both A and B fragments are contiguous v2f.
// One wave computes a (16*MT) x (16*NT) tile with V_WMMA_F32_16X16X4_F32,
// software-pipelined: fragments for k+4 are loaded before the k WMMA pack.
// A-frag (16x4): lanes 0-15 hold M=lane, K={k,k+1}; lanes 16-31 K={k+2,k+3}.
// B-frag (4x16): lanes 0-15 hold N=lane, K={k,k+1}; lanes 16-31 K={k+2,k+3}.
// C/D (16x16):   VGPR r, lanes 0-15 -> (M=r, N=lane); lanes 16-31 -> (M=r+8).
// EPI: 0 = plain store, 1 = softplus(acc + 2*bias[col])
// =====================================================================
template <int MT, int NT, int EPI>
__global__ __launch_bounds__(256)
void wmma_gemm_f32t(const float* __restrict__ A, const float* __restrict__ Bt,
                    const float* __restrict__ bias, float* __restrict__ C,
                    int M, int N, int K, int lda) {
    const int wave   = (blockIdx.x * blockDim.x + threadIdx.x) >> 5;
    const int lane   = threadIdx.x & 31;
    const int tilesN = N / (16 * NT);
    const int tm     = (wave / tilesN) * (16 * MT);
    const int tn     = (wave % tilesN) * (16 * NT);
    if (tm >= M) return;                       // wave-uniform guard

    const int half = lane >> 4;                // 0: K pair {0,1}, 1: {2,3}
    const int l16  = lane & 15;

    v8f acc[MT][NT];
#pragma unroll
    for (int mi = 0; mi < MT; ++mi)
#pragma unroll
        for (int ni = 0; ni < NT; ++ni)
            acc[mi][ni] = (v8f){0.f,0.f,0.f,0.f,0.f,0.f,0.f,0.f};

    const float* __restrict__ ap[MT];
    const float* __restrict__ bp[NT];
#pragma unroll
    for (int mi = 0; mi < MT; ++mi)
        ap[mi] = A + (size_t)(tm + mi * 16 + l16) * lda + half * 2;
#pragma unroll
    for (int ni = 0; ni < NT; ++ni)
        bp[ni] = Bt + (size_t)(tn + ni * 16 + l16) * K + half * 2;

    v2f af[MT], bf[NT];
#pragma unroll
    for (int mi = 0; mi < MT; ++mi) af[mi] = *(const v2f*)(ap[mi]);
#pragma unroll
    for (int ni = 0; ni < NT; ++ni) bf[ni] = *(const v2f*)(bp[ni]);

    int k = 0;
    for (; k + 4 < K; k += 4) {
        v2f afn[MT], bfn[NT];
#pragma unroll
        for (int mi = 0; mi < MT; ++mi) afn[mi] = *(const v2f*)(ap[mi] + k + 4);
#pragma unroll
        for (int ni = 0; ni < NT; ++ni) bfn[ni] = *(const v2f*)(bp[ni] + k + 4);
#pragma unroll
        for (int mi = 0; mi < MT; ++mi)
#pragma unroll
            for (int ni = 0; ni < NT; ++ni)
                acc[mi][ni] = __builtin_amdgcn_wmma_f32_16x16x4_f32(
                    false, af[mi], false, bf[ni], (short)0, acc[mi][ni], false, false);
#pragma unroll
        for (int mi = 0; mi < MT; ++mi) af[mi] = afn[mi];
#pragma unroll
        for (int ni = 0; ni < NT; ++ni) bf[ni] = bfn[ni];
    }
    // final K-step (no prefetch)
#pragma unroll
    for (int mi = 0; mi < MT; ++mi)
#pragma unroll
        for (int ni = 0; ni < NT; ++ni)
            acc[mi][ni] = __builtin_amdgcn_wmma_f32_16x16x4_f32(
                false, af[mi], false, bf[ni], (short)0, acc[mi][ni], false, false);

#pragma unroll
    for (int mi = 0; mi < MT; ++mi)
#pragma unroll
        for (int ni = 0; ni < NT; ++ni) {
            const int col = tn + ni * 16 + l16;
#pragma unroll
            for (int r = 0; r < 8; ++r) {
                const int row = tm + mi * 16 + r + half * 8;
                float v = acc[mi][ni][r];
                if (EPI == 1) {                // delta = softplus(acc + 2*b)
                    float x = v + 2.0f * bias[col];
                    v = (x > 20.0f) ? x : __logf(1.0f + __expf(x));
                }
                C[(size_t)row * N + col] = v;
            }
        }
}

// =====================================================================
// Depthwise shifted causal conv + bias + SiLU.
// Reference pads (2p, p) with p=3 then slices [:L], so out[t] reads
// x[t-6+k], k=0..3. x lives in the first half of the xz buffer (stride 4096).
// =====================================================================
__global__ __launch_bounds__(256)
void conv_silu_kernel(const float* __restrict__ xz, const float* __restrict__ cw,
                      const float* __restrict__ cb, float* __restrict__ out) {
    const int idx = blockIdx.x * blockDim.x + threadIdx.x;   // over BL*D_INNER
    const int d   = idx & (D_INNER - 1);
    const int row = idx >> 11;                               // b*L + t
    const int t   = row & (SEQLEN - 1);

    float s = cb[d];
#pragma unroll
    for (int k = 0; k < D_CONV; ++k) {
        const int tt = t - 6 + k;
        if (tt >= 0)
            s += cw[d * D_CONV + k] * xz[((size_t)(row - t + tt)) * (2 * D_INNER) + d];
    }
    out[idx] = s / (1.0f + __expf(-s));                      // silu
}

// =====================================================================
// Selective scan. One lane per (b, d, n) state: 131072 lanes.
// h_n(t) = h_n(t-1)*exp(delta*A_n) + delta*u*B_n ; y = sum_n h_n * C_n.
// Reduction over the 16 states of a channel via shfl_xor (xor masks
// 1,2,4,8 stay within the lane 16-group).
// =====================================================================
__global__ __launch_bounds__(256)
void scan_kernel(const float* __restrict__ delta, const float* __restrict__ u,
                 const float* __restrict__ xdbl, const float* __restrict__ A_log,
                 float* __restrict__ ys) {
    const int gtid = blockIdx.x * blockDim.x + threadIdx.x;
    const int n = gtid & (D_STATE - 1);
    const int d = (gtid >> 4) & (D_INNER - 1);
    const int b = gtid >> 15;

    const float a = -__expf(A_log[d * D_STATE + n]);
    float h = 0.0f;
    const size_t rowBase = (size_t)b * SEQLEN;

    for (int t = 0; t < SEQLEN; ++t) {
        const size_t row = rowBase + t;
        const float dl = delta[row * D_INNER + d];
        const float uu = u[row * D_INNER + d];
        const float Bn = xdbl[row * XDBL_N + DT_RANK + n];
        const float Cn = xdbl[row * XDBL_N + DT_RANK + D_STATE + n];

        h = h * __expf(dl * a) + dl * uu * Bn;
        float y = h * Cn;
        y += __shfl_xor(y, 1, 32);
        y += __shfl_xor(y, 2, 32);
        y += __shfl_xor(y, 4, 32);
        y += __shfl_xor(y, 8, 32);
        if (n == 0) ys[row * D_INNER + d] = y;
    }
}

// =====================================================================
// y = (ys + x_conv * D_skip) * silu(z);  z = xz[..., 2048 + d]
// =====================================================================
__global__ __launch_bounds__(256)
void combine_kernel(const float* __restrict__ ys, const float* __restrict__ xconv,
                    const float* __restrict__ Dskip, const float* __restrict__ xz,
                    float* __restrict__ out) {
    const int idx = blockIdx.x * blockDim.x + threadIdx.x;
    const int d   = idx & (D_INNER - 1);
    const int row = idx >> 11;
    const float z = xz[(size_t)row * (2 * D_INNER) + D_INNER + d];
    const float y = ys[idx] + xconv[idx] * Dskip[d];
    out[idx] = y * (z / (1.0f + __expf(-z)));
}

// =====================================================================
// Launch
// =====================================================================
extern "C" void kernel_launch(void* const* d_in, const int* in_sizes, int n_in,
                              void* d_out, int out_size, void* d_ws, size_t ws_size,
                              hipStream_t stream) {
    const float* hs        = (const float*)d_in[0];
    const float* in_proj_w = (const float*)d_in[1];
    const float* conv_w    = (const float*)d_in[2];
    const float* conv_b    = (const float*)d_in[3];
    const float* x_proj_w  = (const float*)d_in[4];
    const float* dt_proj_w = (const float*)d_in[5];
    const float* dt_proj_b = (const float*)d_in[6];
    const float* A_log     = (const float*)d_in[7];
    const float* D_skip    = (const float*)d_in[8];
    const float* out_proj  = (const float*)d_in[9];
    float* out = (float*)d_out;

    // workspace carve-up (floats)
    float* ws       = (float*)d_ws;
    float* xz       = ws;                                   // 8192*4096
    float* xconv    = xz      + (size_t)BL * 2 * D_INNER;   // 8192*2048
    float* xdbl     = xconv   + (size_t)BL * D_INNER;       // 8192*96
    float* delta    = xdbl    + (size_t)BL * XDBL_N;        // 8192*2048
    float* ysbuf    = delta   + (size_t)BL * D_INNER;       // 8192*2048
    float* ycomb    = ysbuf   + (size_t)BL * D_INNER;       // 8192*2048
    float* inprojT  = ycomb   + (size_t)BL * D_INNER;       // 4096*1024
    float* xprojT   = inprojT + (size_t)(2 * D_INNER) * D_MODEL;  // 96*2048
    float* dtprojT  = xprojT  + (size_t)XDBL_N * D_INNER;   // 2048*64
    float* outprojT = dtprojT + (size_t)D_INNER * DT_RANK;  // 1024*2048

    const int WPB = 8;   // waves per 256-thread block

    // 0) transpose the four weight matrices (tiny, L2-resident afterwards)
    transpose_kernel<<<(D_MODEL * 2 * D_INNER + 255) / 256, 256, 0, stream>>>(
        in_proj_w, inprojT, D_MODEL, 2 * D_INNER);
    transpose_kernel<<<(D_INNER * XDBL_N + 255) / 256, 256, 0, stream>>>(
        x_proj_w, xprojT, D_INNER, XDBL_N);
    transpose_kernel<<<(DT_RANK * D_INNER + 255) / 256, 256, 0, stream>>>(
        dt_proj_w, dtprojT, DT_RANK, D_INNER);
    transpose_kernel<<<(D_INNER * D_MODEL + 255) / 256, 256, 0, stream>>>(
        out_proj, outprojT, D_INNER, D_MODEL);

    // 1) xz = hs @ in_proj_w        (8192 x 4096 x 1024), 32x64 per wave
    {
        int waves = (BL / 32) * (2 * D_INNER / 64);
        hipLaunchKernelGGL((wmma_gemm_f32t<2, 4, 0>), dim3(waves / WPB), dim3(256), 0, stream,
                           hs, inprojT, nullptr, xz, BL, 2 * D_INNER, D_MODEL, D_MODEL);
    }
    // 2) depthwise conv + SiLU
    conv_silu_kernel<<<(BL * D_INNER) / 256, 256, 0, stream>>>(xz, conv_w, conv_b, xconv);

    // 3) x_dbl = x_conv @ x_proj_w  (8192 x 96 x 2048), 32x96 per wave
    {
        int waves = (BL / 32) * (XDBL_N / 96);
        hipLaunchKernelGGL((wmma_gemm_f32t<2, 6, 0>), dim3(waves / WPB), dim3(256), 0, stream,
                           xconv, xprojT, nullptr, xdbl, BL, XDBL_N, D_INNER, D_INNER);
    }
    // 4) delta = softplus(dt_low @ dt_proj_w + 2*b)  (8192 x 2048 x 64, lda=96)
    {
        int waves = (BL / 32) * (D_INNER / 64);
        hipLaunchKernelGGL((wmma_gemm_f32t<2, 4, 1>), dim3(waves / WPB), dim3(256), 0, stream,
                           xdbl, dtprojT, dt_proj_b, delta, BL, D_INNER, DT_RANK, XDBL_N);
    }
    // 5) selective scan  (131072 lanes)
    scan_kernel<<<(BATCH * D_INNER * D_STATE) / 256, 256, 0, stream>>>(
        delta, xconv, xdbl, A_log, ysbuf);

    // 6) combine
    combine_kernel<<<(BL * D_INNER) / 256, 256, 0, stream>>>(ysbuf, xconv, D_skip, xz, ycomb);

    // 7) out = ycomb @ out_proj_w   (8192 x 1024 x 2048)
    {
        int waves = (BL / 32) * (D_MODEL / 64);
        hipLaunchKernelGGL((wmma_gemm_f32t<2, 4, 0>), dim3(waves / WPB), dim3(256), 0, stream,
                           ycomb, outprojT, nullptr, out, BL, D_MODEL, D_INNER, D_INNER);
    }
}